// ContDecoder_48833778156096
// MI455X (gfx1250) — compile-verified
//
#include <hip/hip_runtime.h>

// ---------------- problem constants (match reference) ----------------
#define B_      8
#define N_      32768
#define CTX_C   64
#define HC      64          // context grid H=W
#define HL      32          // lr grid H=W
#define HID     512
#define TILE_P  32          // points per workgroup
#define XSTRIDE 516         // LDS stride for hidden activations (bank spread)
#define ISTRIDE 84          // LDS stride for input features (69 used, pad->80)
#define WPRS    1056        // floats per staged weight pair-row (512*2 + 32 pad)

typedef float v2f __attribute__((ext_vector_type(2)));
typedef float v8f __attribute__((ext_vector_type(8)));

__device__ __forceinline__ float corner(const float* __restrict__ img, int base,
                                        int Hh, int Ww, int ix, int iy) {
  if (ix < 0 || ix >= Ww || iy < 0 || iy >= Hh) return 0.0f;
  return img[base + iy * Ww + ix];
}

// One MLP layer: out(512) = ReLU(A @ W + b), A = [sX(512) | sI(69..)] if CONCAT
// else A = sI alone. Result written back into sX. Whole workgroup cooperates.
// Wave tiling: 8 waves = 2 M-subtiles(16 rows) x 4 N-groups(128 cols).
// Weight K-chunks (16 rows) are register-pipelined: chunk ch+1 is loaded from
// global (L2-resident) into VGPRs while chunk ch's 32 WMMAs execute.
template <bool CONCAT>
__device__ __forceinline__ void mlp_layer(const float* __restrict__ Wg,
                                          const float* __restrict__ bg,
                                          int Kact, int nchunks,
                                          float* __restrict__ sX,
                                          const float* __restrict__ sI,
                                          float* __restrict__ sW) {
  const int tid  = threadIdx.x;
  const int lane = tid & 31;
  const int wave = tid >> 5;
  const int m0   = (wave >> 2) * 16;   // M subtile base row
  const int c0   = (wave & 3) * 128;   // N group base col
  const int colb = lane & 15;
  const int kh   = lane >> 4;          // lane half (K offset selector)
  const int rowA = m0 + colb;          // A-matrix row held by this lane

  v8f acc[8];
#pragma unroll
  for (int t = 0; t < 8; ++t) {
    float bb = bg[c0 + 16 * t + colb]; // bias depends only on column
#pragma unroll
    for (int i = 0; i < 8; ++i) acc[t][i] = bb;
  }

  // Prefetch registers: 4 pair-slots/thread, each = rows (2pr, 2pr+1) x 4 cols.
  // Slot s = i*256 + tid : pr = s>>7 (0..7), c4 = s&127 (float4 column).
  float4 pfE[4], pfO[4];

  auto prefetch = [&](int k0) {
#pragma unroll
    for (int i = 0; i < 4; ++i) {
      int s  = i * 256 + tid;
      int pr = s >> 7;
      int c4 = s & 127;
      int gkE = k0 + 2 * pr;
      pfE[i] = make_float4(0.f, 0.f, 0.f, 0.f);
      pfO[i] = make_float4(0.f, 0.f, 0.f, 0.f);
      if (gkE < Kact)     pfE[i] = *(const float4*)(Wg + gkE * HID + c4 * 4);
      if (gkE + 1 < Kact) pfO[i] = *(const float4*)(Wg + (gkE + 1) * HID + c4 * 4);
    }
  };

  prefetch(0);

  for (int ch = 0; ch < nchunks; ++ch) {
    const int k0 = ch * 16;
    // ---- commit prefetched chunk to LDS (pair-interleaved, b128 stores) ----
#pragma unroll
    for (int i = 0; i < 4; ++i) {
      int s  = i * 256 + tid;
      int pr = s >> 7;
      int c4 = s & 127;
      int base = pr * WPRS + c4 * 8;   // byte addr multiple of 16 -> b128 ok
      float4 lo = make_float4(pfE[i].x, pfO[i].x, pfE[i].y, pfO[i].y);
      float4 hi = make_float4(pfE[i].z, pfO[i].z, pfE[i].w, pfO[i].w);
      *(float4*)&sW[base + 0] = lo;
      *(float4*)&sW[base + 4] = hi;
    }
    __syncthreads();
    // ---- issue next chunk's global loads; latency hides behind WMMAs ----
    if (ch + 1 < nchunks) prefetch(k0 + 16);
    // ---- 4 WMMA K-steps of 4 over this chunk ----
#pragma unroll
    for (int kk = 0; kk < 16; kk += 4) {
      const int kg = k0 + kk;
      const float* ap;
      if (CONCAT && kg < HID)
        ap = &sX[rowA * XSTRIDE + kg + 2 * kh];
      else
        ap = &sI[rowA * ISTRIDE + (CONCAT ? (kg - HID) : kg) + 2 * kh];
      v2f a = *(const v2f*)ap;  // A 16x4 frag: lane half selects K pair {0,1}/{2,3}
      const float* wb = &sW[((kk >> 1) + kh) * WPRS + (c0 + colb) * 2];
#pragma unroll
      for (int t = 0; t < 8; ++t) {
        v2f b = *(const v2f*)(wb + 32 * t);  // B 4x16 frag, same pair layout
        acc[t] = __builtin_amdgcn_wmma_f32_16x16x4_f32(
            false, a, false, b, (short)0, acc[t], false, false);
      }
    }
    __syncthreads();  // all waves done reading sW (and sX) before next commit
  }
  // ---- ReLU + writeback (all reads of sX finished at the barrier above) ----
#pragma unroll
  for (int t = 0; t < 8; ++t) {
#pragma unroll
    for (int i = 0; i < 8; ++i) {
      int row = m0 + kh * 8 + i;       // C/D layout: VGPR i -> row i / i+8 by half
      sX[row * XSTRIDE + c0 + 16 * t + colb] = fmaxf(acc[t][i], 0.0f);
    }
  }
  __syncthreads();
}

__global__ void __launch_bounds__(256)
contdec_kernel(const float* __restrict__ lr_grid, const float* __restrict__ ctx_grid,
               const float* __restrict__ coord,
               const float* __restrict__ W0, const float* __restrict__ b0,
               const float* __restrict__ W1, const float* __restrict__ b1,
               const float* __restrict__ W2, const float* __restrict__ b2,
               const float* __restrict__ W3, const float* __restrict__ b3,
               float* __restrict__ out) {
  extern __shared__ float smem[];
  float* sX = smem;                         // [32][516] hidden activations
  float* sI = sX + TILE_P * XSTRIDE;        // [32][84]  input features (padded)
  float* sW = sI + TILE_P * ISTRIDE;        // [8][1056] staged weight chunk

  const int tid = threadIdx.x;

  // ---------------- phase 0: grid-sample features into sI ----------------
  {
    int p   = tid & 31;       // point within tile
    int sub = tid >> 5;       // 8 helpers per point
    int q   = blockIdx.x * TILE_P + p;
    int b   = q >> 15;        // N_ = 2^15
    int n   = q & (N_ - 1);
    float cx = coord[(b * N_ + n) * 2 + 0];
    float cy = coord[(b * N_ + n) * 2 + 1];

    // context grid (64x64), align_corners=False
    float fx = (cx + 1.f) * (HC * 0.5f) - 0.5f;
    float fy = (cy + 1.f) * (HC * 0.5f) - 0.5f;
    float fx0 = floorf(fx), fy0 = floorf(fy);
    int   x0 = (int)fx0,    y0 = (int)fy0;
    float wx = fx - fx0,    wy = fy - fy0;
    float w00 = (1.f - wx) * (1.f - wy), w10 = wx * (1.f - wy);
    float w01 = (1.f - wx) * wy,         w11 = wx * wy;
    for (int c = sub; c < CTX_C; c += 8) {
      int base = ((b * CTX_C + c) * HC) * HC;
      float v = w00 * corner(ctx_grid, base, HC, HC, x0,     y0)
              + w10 * corner(ctx_grid, base, HC, HC, x0 + 1, y0)
              + w01 * corner(ctx_grid, base, HC, HC, x0,     y0 + 1)
              + w11 * corner(ctx_grid, base, HC, HC, x0 + 1, y0 + 1);
      sI[p * ISTRIDE + c] = v;
    }
    if (sub == 0) {
      sI[p * ISTRIDE + 64] = cx;
      sI[p * ISTRIDE + 65] = cy;
      float gx = (cx + 1.f) * (HL * 0.5f) - 0.5f;
      float gy = (cy + 1.f) * (HL * 0.5f) - 0.5f;
      float gx0 = floorf(gx), gy0 = floorf(gy);
      int   lx0 = (int)gx0,   ly0 = (int)gy0;
      float vx = gx - gx0,    vy = gy - gy0;
      float u00 = (1.f - vx) * (1.f - vy), u10 = vx * (1.f - vy);
      float u01 = (1.f - vx) * vy,          u11 = vx * vy;
      for (int c = 0; c < 3; ++c) {
        int base = ((b * 3 + c) * HL) * HL;
        float v = u00 * corner(lr_grid, base, HL, HL, lx0,     ly0)
                + u10 * corner(lr_grid, base, HL, HL, lx0 + 1, ly0)
                + u01 * corner(lr_grid, base, HL, HL, lx0,     ly0 + 1)
                + u11 * corner(lr_grid, base, HL, HL, lx0 + 1, ly0 + 1);
        sI[p * ISTRIDE + 66 + c] = v;
      }
      for (int c = 69; c < 80; ++c) sI[p * ISTRIDE + c] = 0.f;  // K padding
    }
  }
  __syncthreads();

  // ---------------- fused MLP: all matmuls via V_WMMA_F32_16X16X4_F32 ----
  mlp_layer<false>(W0, b0,  69,  5, sX, sI, sW);   // K padded 69 -> 80
  mlp_layer<true >(W1, b1, 581, 37, sX, sI, sW);   // K padded 581 -> 592
  mlp_layer<true >(W2, b2, 581, 37, sX, sI, sW);

  // ---------------- final 512 -> 2 (memory-trivial, VALU + shuffles) ----
  ((float4*)sW)[tid] = ((const float4*)W3)[tid];   // stage all 1024 floats of W3
  __syncthreads();
  {
    int p = tid >> 3, sub = tid & 7;               // 8 partial sums per point
    float s0 = 0.f, s1 = 0.f;
    int kb = sub * 64;
    for (int k = 0; k < 64; ++k) {
      float xv = sX[p * XSTRIDE + kb + k];
      s0 += xv * sW[(kb + k) * 2 + 0];
      s1 += xv * sW[(kb + k) * 2 + 1];
    }
    for (int o = 4; o; o >>= 1) {                  // reduce within groups of 8 lanes
      s0 += __shfl_xor(s0, o, 8);
      s1 += __shfl_xor(s1, o, 8);
    }
    if (sub == 0) {
      int q = blockIdx.x * TILE_P + p;
      out[q * 2 + 0] = s0 + b3[0];
      out[q * 2 + 1] = s1 + b3[1];
    }
  }
}

extern "C" void kernel_launch(void* const* d_in, const int* in_sizes, int n_in,
                              void* d_out, int out_size, void* d_ws, size_t ws_size,
                              hipStream_t stream) {
  (void)in_sizes; (void)n_in; (void)out_size; (void)d_ws; (void)ws_size;
  const float* lr  = (const float*)d_in[0];
  const float* ctx = (const float*)d_in[1];
  const float* co  = (const float*)d_in[2];
  const float* W0  = (const float*)d_in[3];
  const float* b0  = (const float*)d_in[4];
  const float* W1  = (const float*)d_in[5];
  const float* b1  = (const float*)d_in[6];
  const float* W2  = (const float*)d_in[7];
  const float* b2  = (const float*)d_in[8];
  const float* W3  = (const float*)d_in[9];
  const float* b3  = (const float*)d_in[10];

  const int nblk = (B_ * N_) / TILE_P;  // 8192 workgroups
  const size_t smem_bytes =
      (size_t)(TILE_P * XSTRIDE + TILE_P * ISTRIDE + 8 * WPRS) * sizeof(float); // ~110.6 KB

  contdec_kernel<<<nblk, 256, smem_bytes, stream>>>(
      lr, ctx, co, W0, b0, W1, b1, W2, b2, W3, b3, (float*)d_out);
}